// BillehColumn_14568529068195
// MI455X (gfx1250) — compile-verified
//
#include <hip/hip_runtime.h>
#include <stdint.h>

// Streaming vector types for non-temporal loads.
typedef int   v4i __attribute__((ext_vector_type(4)));
typedef float v2f __attribute__((ext_vector_type(2)));

// ---------------------------------------------------------------------------
// Kernel 1: zero the output accumulator (d_out is poisoned by the harness and
// must be re-zeroed every launch for deterministic atomic accumulation).
// ---------------------------------------------------------------------------
__global__ void zero_f32(float* __restrict__ p, int n) {
    int i = blockIdx.x * blockDim.x + threadIdx.x;
    if (i < n) p[i] = 0.0f;
}

// ---------------------------------------------------------------------------
// Kernel 2: compress the binary spike buffer z[2][n_post] into an interleaved
// bitmask: masks[w] = { bits for batch0 neurons w*32..w*32+31,
//                       bits for batch1 neurons w*32..w*32+31 }.
// 2 * 50000 bits -> ~12.5 KB, rebuilt in d_ws every call.
// ---------------------------------------------------------------------------
__global__ void build_masks(const float* __restrict__ z,
                            uint2* __restrict__ masks,
                            int n_post, int w_pad) {
    int w = blockIdx.x * blockDim.x + threadIdx.x;
    if (w >= w_pad) return;
    uint32_t m0 = 0u, m1 = 0u;
    int base = w << 5;
    for (int j = 0; j < 32; ++j) {
        int idx = base + j;
        if (idx < n_post) {
            if (z[idx]          != 0.0f) m0 |= (1u << j);
            if (z[n_post + idx] != 0.0f) m1 |= (1u << j);
        }
    }
    masks[w] = make_uint2(m0, m1);
}

// ---------------------------------------------------------------------------
// Kernel 3: stream the synapse list (NT loads), test spikes against the LDS
// bitmask (staged via gfx1250 async global->LDS copy), scatter weights with
// f32 global atomics. Spikes are exactly 0.0/1.0, so w*z == w when the bit
// is set.
// ---------------------------------------------------------------------------
#define SMASK_WORDS 1600   // >= ceil(50000/32) rounded up; 12.8 KB of LDS

__global__ __launch_bounds__(256) void scatter_currents(
        const v4i*  __restrict__ idx2,     // 2 synapses per element: post0,pre0,post1,pre1
        const v2f*  __restrict__ w2,       // 2 weights per element
        const uint2* __restrict__ masks_g, // interleaved spike bitmask in d_ws
        float* __restrict__ out,           // [2 * n_post]
        int n_pairs, int n_post, int w_pad) {
    __shared__ uint2 smask[SMASK_WORDS];

    // --- CDNA5 async copy: global -> LDS, tracked by ASYNCcnt ---------------
    // Generic pointers to LDS carry the LDS byte offset in their low 32 bits.
    const uint32_t lds_base = (uint32_t)(uintptr_t)(&smask[0]);
    const uint32_t nbytes   = (uint32_t)w_pad * 8u;          // uint2 = 8 B/word
    for (uint32_t off = threadIdx.x * 16u; off < nbytes; off += blockDim.x * 16u) {
        uint32_t laddr = lds_base + off;
        asm volatile("global_load_async_to_lds_b128 %0, %1, %2"
                     :
                     : "v"(laddr), "v"(off), "s"(masks_g)
                     : "memory");
    }
    asm volatile("s_wait_asynccnt 0" ::: "memory");
    __syncthreads();

    float* out1 = out + n_post;
    const int stride = gridDim.x * blockDim.x;
    for (int i = blockIdx.x * blockDim.x + threadIdx.x; i < n_pairs; i += stride) {
        v4i pp = __builtin_nontemporal_load(&idx2[i]);   // b128, TH=NT
        v2f ww = __builtin_nontemporal_load(&w2[i]);     // b64,  TH=NT
        // synapse 2i: post = pp.x, pre = pp.y, weight = ww.x
        {
            uint2    m   = smask[((uint32_t)pp.y) >> 5]; // ds_load_b64
            uint32_t bit = 1u << (((uint32_t)pp.y) & 31u);
            if (m.x & bit) atomicAdd(&out [pp.x], ww.x);
            if (m.y & bit) atomicAdd(&out1[pp.x], ww.x);
        }
        // synapse 2i+1: post = pp.z, pre = pp.w, weight = ww.y
        {
            uint2    m   = smask[((uint32_t)pp.w) >> 5];
            uint32_t bit = 1u << (((uint32_t)pp.w) & 31u);
            if (m.x & bit) atomicAdd(&out [pp.z], ww.y);
            if (m.y & bit) atomicAdd(&out1[pp.z], ww.y);
        }
    }
}

// ---------------------------------------------------------------------------
// Launch wrapper.
//   d_in[0] rec_z_buf       float32 [2 * n_post]
//   d_in[1] synapse_indices int32   [n_syn * 2]  (post, pre interleaved)
//   d_in[2] weight_values   float32 [n_syn]
//   d_in[3] n_post_neurons  (device scalar; n_post derived from in_sizes)
// ---------------------------------------------------------------------------
extern "C" void kernel_launch(void* const* d_in, const int* in_sizes, int n_in,
                              void* d_out, int out_size, void* d_ws, size_t ws_size,
                              hipStream_t stream) {
    const float* rec_z = (const float*)d_in[0];
    const int*   syn   = (const int*)  d_in[1];
    const float* wv    = (const float*)d_in[2];

    const int n_syn   = in_sizes[2];          // 10,000,000
    const int n_post  = in_sizes[0] / 2;      // B == 2 -> 50,000
    const int n_pairs = n_syn / 2;            // even by construction
    const int w_words = (n_post + 31) / 32;
    const int w_pad   = (w_words + 3) & ~3;   // 16-byte multiple for b128 copy

    uint2* masks = (uint2*)d_ws;              // 12,512 B of scratch
    float* out   = (float*)d_out;

    zero_f32<<<(out_size + 255) / 256, 256, 0, stream>>>(out, out_size);
    build_masks<<<(w_pad + 255) / 256, 256, 0, stream>>>(rec_z, masks, n_post, w_pad);

    const int blocks = 2048;                  // ~9.5 synapse-pairs per thread
    scatter_currents<<<blocks, 256, 0, stream>>>(
        (const v4i*)syn, (const v2f*)wv, masks, out, n_pairs, n_post, w_pad);
}